// VimBlock_68298569941703
// MI455X (gfx1250) — compile-verified
//
#include <hip/hip_runtime.h>
#include <hip/hip_bf16.h>

// ---------------- problem constants (from the reference) ----------------
#define BATCH   4
#define DMODEL  96
#define LSEQ    4096            // 64*64
#define MROWS   (BATCH*LSEQ)    // 16384
#define ED      192             // d_inner
#define NPROJ   384             // 2*d_inner
#define RDT     6               // dt_rank
#define NST     16              // d_state
#define DBCN    (RDT + 2*NST)   // 38
#define KCONV   4
#define NCHUNK  64
#define CLEN    (LSEQ/NCHUNK)   // 64
#define EPSF    1e-5f

typedef float v2f __attribute__((ext_vector_type(2)));
typedef float v8f __attribute__((ext_vector_type(8)));

__device__ __forceinline__ float silu_f(float v) { return v / (1.0f + __expf(-v)); }
__device__ __forceinline__ float softplus_f(float v) {
    return (v > 20.0f) ? v : log1pf(__expf(v));
}

// fp32 WMMA: D(16x16) = A(16x4) * B(4x16) + C.  Full reference precision.
__device__ __forceinline__ v8f wmma_f32_k4(v2f a, v2f b, v8f c) {
    return __builtin_amdgcn_wmma_f32_16x16x4_f32(
        /*neg_a=*/false, a, /*neg_b=*/false, b,
        /*c_mod=*/(short)0, c, /*reuse_a=*/false, /*reuse_b=*/false);
}

// ---- CDNA5 async global->LDS copies (ASYNCcnt-tracked, no VGPR round-trip) ----
__device__ __forceinline__ unsigned lds_off_of(const void* p) {
    return (unsigned)(size_t)p;           // __shared__ pointer == LDS byte offset
}
__device__ __forceinline__ void async_g2l_b128(unsigned lds_off, const float* gsrc) {
    asm volatile("global_load_async_to_lds_b128 %0, %1, off"
                 :: "v"(lds_off), "v"(gsrc) : "memory");
}
__device__ __forceinline__ void async_g2l_b32(unsigned lds_off, const float* gsrc) {
    asm volatile("global_load_async_to_lds_b32 %0, %1, off"
                 :: "v"(lds_off), "v"(gsrc) : "memory");
}
__device__ __forceinline__ void wait_async_all() {
    asm volatile("s_wait_asynccnt 0x0" ::: "memory");
}

// =====================================================================
// K0: RMSNorm over d_model + input projection (16384x96) x (96x384)
//     one block = one 16-row M tile; 4 waves, each wave owns 6 N tiles.
//     Gathered (transposing) tile load goes straight to LDS with
//     global_load_async_to_lds_b32.
// =====================================================================
__global__ void k_rms_inproj(const float* __restrict__ x,      // (B,96,4096)
                             const float* __restrict__ W_in,   // (384,96)
                             const float* __restrict__ norm_w, // (96)
                             float* __restrict__ xz)           // (M,384)
{
    __shared__ float sA[16][100];     // padded vs. 64 banks
    __shared__ float sScale[16];

    const int tid   = threadIdx.x;
    const int row0  = blockIdx.x * 16;
    const int b     = row0 / LSEQ;
    const int l0    = row0 % LSEQ;

    // async gather: u[row][c] = x[b, c, l0+row]   (1536 tasks, 12 per thread)
    for (int idx = tid; idx < 16 * DMODEL; idx += 128) {
        int r = idx / DMODEL, c = idx % DMODEL;
        async_g2l_b32(lds_off_of(&sA[r][c]),
                      &x[((size_t)b * DMODEL + c) * LSEQ + (l0 + r)]);
    }
    wait_async_all();
    __syncthreads();

    if (tid < 16) {
        float s = 0.f;
        #pragma unroll
        for (int c = 0; c < DMODEL; ++c) { float v = sA[tid][c]; s += v * v; }
        sScale[tid] = rsqrtf(s * (1.0f / DMODEL) + EPSF);
    }
    __syncthreads();

    for (int idx = tid; idx < 16 * DMODEL; idx += 128) {
        int r = idx / DMODEL, c = idx % DMODEL;
        sA[r][c] = sA[r][c] * sScale[r] * norm_w[c];
    }
    __syncthreads();

    const int wave = tid >> 5;
    const int lane = tid & 31;
    const int arow = lane & 15;
    const int khi  = (lane >= 16) ? 2 : 0;       // lanes 16..31 hold K+2 (ISA A layout)

    for (int nt = wave; nt < NPROJ / 16; nt += 4) {
        const int n = nt * 16 + (lane & 15);
        v8f acc = {0.f,0.f,0.f,0.f,0.f,0.f,0.f,0.f};
        #pragma unroll
        for (int ks = 0; ks < DMODEL / 4; ++ks) {
            const int kb = ks * 4 + khi;
            v2f a; a.x = sA[arow][kb];            a.y = sA[arow][kb + 1];
            v2f bb; bb.x = W_in[(size_t)n * DMODEL + kb];
                    bb.y = W_in[(size_t)n * DMODEL + kb + 1];
            acc = wmma_f32_k4(a, bb, acc);
        }
        // C/D layout: VGPR i -> row i (lanes 0-15) / row i+8 (lanes 16-31)
        #pragma unroll
        for (int i = 0; i < 8; ++i) {
            int r = i + ((lane >= 16) ? 8 : 0);
            xz[(size_t)(row0 + r) * NPROJ + n] = acc[i];
        }
    }
}

// =====================================================================
// K1: causal depthwise conv (K=4) + bias + SiLU  -> xs (M,192)
// =====================================================================
__global__ void k_conv_silu(const float* __restrict__ xz,
                            const float* __restrict__ conv_w, // (192,1,4)
                            const float* __restrict__ conv_b, // (192)
                            float* __restrict__ xs)           // (M,192)
{
    int t = blockIdx.x * blockDim.x + threadIdx.x;
    if (t >= MROWS * ED) return;
    int e = t % ED;
    int row = t / ED;
    int l = row % LSEQ;
    float acc = conv_b[e];
    #pragma unroll
    for (int k = 0; k < KCONV; ++k) {
        int ll = l - (KCONV - 1) + k;
        if (ll >= 0)
            acc += conv_w[e * KCONV + k] * xz[(size_t)(row - (KCONV - 1) + k) * NPROJ + e];
    }
    xs[(size_t)row * ED + e] = silu_f(acc);
}

// =====================================================================
// K2: x-projection (192->38), then delta = softplus(dbc[:6] @ W_dt.T + b_dt)
//     one thread = one row; 64x192 tile staged with async b128 copies.
// =====================================================================
__global__ void k_xproj(const float* __restrict__ xs,
                        const float* __restrict__ W_x,   // (38,192)
                        const float* __restrict__ W_dt,  // (192,6)
                        const float* __restrict__ b_dt,  // (192)
                        float* __restrict__ delta,       // (M,192)
                        float* __restrict__ Bm,          // (M,16)
                        float* __restrict__ Cm)          // (M,16)
{
    __shared__ float sX[64][193];   // 193 stride -> conflict-free column walks
    const int tid = threadIdx.x;
    const int row0 = blockIdx.x * 64;

    // 64 rows x 48 b128-chunks = 3072 tasks, 48 per thread (exact, no tails)
    for (int idx = tid; idx < 64 * (ED / 4); idx += 64) {
        int r = idx / (ED / 4), ch = idx % (ED / 4);
        async_g2l_b128(lds_off_of(&sX[r][ch * 4]),
                       &xs[(size_t)(row0 + r) * ED + ch * 4]);
    }
    wait_async_all();
    __syncthreads();

    const int row = row0 + tid;
    float dbc[DBCN];
    #pragma unroll 2
    for (int j = 0; j < DBCN; ++j) {
        float s = 0.f;
        #pragma unroll 8
        for (int k = 0; k < ED; ++k) s += sX[tid][k] * W_x[(size_t)j * ED + k];
        dbc[j] = s;
    }
    #pragma unroll
    for (int n = 0; n < NST; ++n) {
        Bm[(size_t)row * NST + n] = dbc[RDT + n];
        Cm[(size_t)row * NST + n] = dbc[RDT + NST + n];
    }
    for (int e = 0; e < ED; ++e) {
        float s = b_dt[e];
        #pragma unroll
        for (int j = 0; j < RDT; ++j) s += dbc[j] * W_dt[e * RDT + j];
        delta[(size_t)row * ED + e] = softplus_f(s);
    }
}

// =====================================================================
// K3: chunked scan, pass 1: per-(b,e,n,chunk) compute P = prod(dA),
//     Q = scan(dBx) with h0 = 0.  786432 independent tasks.
// =====================================================================
__global__ void k_scan_chunks(const float* __restrict__ delta,
                              const float* __restrict__ xs,
                              const float* __restrict__ Bm,
                              const float* __restrict__ A_log, // (192,16)
                              float* __restrict__ Pc,          // (12288,64)
                              float* __restrict__ Qc)
{
    int g = blockIdx.x * blockDim.x + threadIdx.x;   // ((b*192+e)*64+chunk)*16+n
    if (g >= BATCH * ED * NST * NCHUNK) return;
    const int n     = g & 15;
    const int chunk = (g >> 4) & (NCHUNK - 1);
    const int e     = (g >> 10) % ED;
    const int b     = g / (16 * NCHUNK * ED);

    const float A = -__expf(A_log[e * NST + n]);
    const int rowbase = b * LSEQ + chunk * CLEN;
    float P = 1.0f, Q = 0.0f;
    for (int l = 0; l < CLEN; ++l) {
        const int row = rowbase + l;
        if (l + 8 < CLEN)
            __builtin_prefetch(&delta[(size_t)(row + 8) * ED + e], 0, 1); // global_prefetch_b8
        const float d  = delta[(size_t)row * ED + e];
        const float a  = __expf(d * A);
        const float bx = d * xs[(size_t)row * ED + e] * Bm[(size_t)row * NST + n];
        Q = a * Q + bx;
        P *= a;
    }
    const size_t lin = ((size_t)(b * ED + e) * NST + n) * NCHUNK + chunk;
    Pc[lin] = P; Qc[lin] = Q;
}

// =====================================================================
// K4: pass 2: sequentially combine the 64 chunk summaries per recurrence
// =====================================================================
__global__ void k_scan_combine(const float* __restrict__ Pc,
                               const float* __restrict__ Qc,
                               float* __restrict__ H0)          // (12288,64)
{
    int t = blockIdx.x * blockDim.x + threadIdx.x;
    if (t >= BATCH * ED * NST) return;
    float h = 0.f;
    const size_t base = (size_t)t * NCHUNK;
    for (int c = 0; c < NCHUNK; ++c) {
        H0[base + c] = h;
        h = Pc[base + c] * h + Qc[base + c];
    }
}

// =====================================================================
// K5: pass 3: replay chunks with correct h0, y = C.h reduced over n with
//     16-lane __shfl_xor (wave32), then y = (y + D*xs) * silu(z)
// =====================================================================
__global__ void k_scan_y(const float* __restrict__ delta,
                         const float* __restrict__ xs,
                         const float* __restrict__ Bm,
                         const float* __restrict__ Cm,
                         const float* __restrict__ A_log,
                         const float* __restrict__ H0,
                         const float* __restrict__ xz,       // z = xz[:,192:]
                         const float* __restrict__ D_param,
                         float* __restrict__ ycore)          // (M,192)
{
    int g = blockIdx.x * blockDim.x + threadIdx.x;
    if (g >= BATCH * ED * NST * NCHUNK) return;
    const int n     = g & 15;
    const int chunk = (g >> 4) & (NCHUNK - 1);
    const int e     = (g >> 10) % ED;
    const int b     = g / (16 * NCHUNK * ED);

    const float A  = -__expf(A_log[e * NST + n]);
    const float Dp = D_param[e];
    float h = H0[((size_t)(b * ED + e) * NST + n) * NCHUNK + chunk];
    const int rowbase = b * LSEQ + chunk * CLEN;

    for (int l = 0; l < CLEN; ++l) {
        const int row = rowbase + l;
        const float d  = delta[(size_t)row * ED + e];
        const float a  = __expf(d * A);
        const float xv = xs[(size_t)row * ED + e];
        h = a * h + d * xv * Bm[(size_t)row * NST + n];
        float p = h * Cm[(size_t)row * NST + n];
        // reduce over the 16 state lanes (stay inside wave32)
        p += __shfl_xor(p, 8, 16);
        p += __shfl_xor(p, 4, 16);
        p += __shfl_xor(p, 2, 16);
        p += __shfl_xor(p, 1, 16);
        if (n == 0) {
            const float zv = xz[(size_t)row * NPROJ + ED + e];
            ycore[(size_t)row * ED + e] = (p + Dp * xv) * silu_f(zv);
        }
    }
}

// =====================================================================
// K6: output projection (16384x192) x (192x96) + residual u, fp32 WMMA.
//     contiguous 16x192 tile staged with async b128; 2 waves, 3 N tiles each.
// =====================================================================
__global__ void k_outproj(const float* __restrict__ ycore,
                          const float* __restrict__ W_out,  // (96,192)
                          const float* __restrict__ x,      // residual source
                          float* __restrict__ out)          // (B,96,4096)
{
    __shared__ float sY[16][200];
    const int tid  = threadIdx.x;
    const int row0 = blockIdx.x * 16;

    // 16 rows x 48 b128-chunks = 768 tasks, 12 per thread (exact)
    for (int idx = tid; idx < 16 * (ED / 4); idx += 64) {
        int r = idx / (ED / 4), ch = idx % (ED / 4);
        async_g2l_b128(lds_off_of(&sY[r][ch * 4]),
                       &ycore[(size_t)(row0 + r) * ED + ch * 4]);
    }
    wait_async_all();
    __syncthreads();

    const int wave = tid >> 5;
    const int lane = tid & 31;
    const int arow = lane & 15;
    const int khi  = (lane >= 16) ? 2 : 0;

    for (int nt = wave; nt < DMODEL / 16; nt += 2) {
        const int c = nt * 16 + (lane & 15);
        v8f acc = {0.f,0.f,0.f,0.f,0.f,0.f,0.f,0.f};
        #pragma unroll
        for (int ks = 0; ks < ED / 4; ++ks) {
            const int kb = ks * 4 + khi;
            v2f a; a.x = sY[arow][kb];            a.y = sY[arow][kb + 1];
            v2f bb; bb.x = W_out[(size_t)c * ED + kb];
                    bb.y = W_out[(size_t)c * ED + kb + 1];
            acc = wmma_f32_k4(a, bb, acc);
        }
        #pragma unroll
        for (int i = 0; i < 8; ++i) {
            int r   = i + ((lane >= 16) ? 8 : 0);
            int row = row0 + r;
            int l   = row & (LSEQ - 1);
            int b   = row >> 12;
            size_t o = ((size_t)b * DMODEL + c) * LSEQ + l;
            out[o] = acc[i] + x[o];
        }
    }
}

// =====================================================================
extern "C" void kernel_launch(void* const* d_in, const int* in_sizes, int n_in,
                              void* d_out, int out_size, void* d_ws, size_t ws_size,
                              hipStream_t stream) {
    const float* x      = (const float*)d_in[0];
    const float* W_in   = (const float*)d_in[1];
    const float* conv_w = (const float*)d_in[2];
    const float* conv_b = (const float*)d_in[3];
    const float* W_x    = (const float*)d_in[4];
    const float* W_dt   = (const float*)d_in[5];
    const float* b_dt   = (const float*)d_in[6];
    const float* A_log  = (const float*)d_in[7];
    const float* D_par  = (const float*)d_in[8];
    const float* W_out  = (const float*)d_in[9];
    const float* norm_w = (const float*)d_in[10];
    float* out = (float*)d_out;

    // workspace layout (floats) — ~75 MB total
    float* ws    = (float*)d_ws;
    float* xz    = ws;                         // M*384
    float* xs    = xz    + (size_t)MROWS * NPROJ;
    float* delta = xs    + (size_t)MROWS * ED;
    float* Bm    = delta + (size_t)MROWS * ED;
    float* Cm    = Bm    + (size_t)MROWS * NST;
    float* Pc    = Cm    + (size_t)MROWS * NST;
    float* Qc    = Pc    + (size_t)BATCH * ED * NST * NCHUNK;
    float* H0    = Qc    + (size_t)BATCH * ED * NST * NCHUNK;
    float* ycore = H0    + (size_t)BATCH * ED * NST * NCHUNK;

    k_rms_inproj<<<MROWS / 16, 128, 0, stream>>>(x, W_in, norm_w, xz);

    {
        int total = MROWS * ED;
        k_conv_silu<<<(total + 255) / 256, 256, 0, stream>>>(xz, conv_w, conv_b, xs);
    }

    k_xproj<<<MROWS / 64, 64, 0, stream>>>(xs, W_x, W_dt, b_dt, delta, Bm, Cm);

    {
        int total = BATCH * ED * NST * NCHUNK;  // 786432
        k_scan_chunks<<<(total + 255) / 256, 256, 0, stream>>>(delta, xs, Bm, A_log, Pc, Qc);
        int rec = BATCH * ED * NST;             // 12288
        k_scan_combine<<<(rec + 255) / 256, 256, 0, stream>>>(Pc, Qc, H0);
        k_scan_y<<<(total + 255) / 256, 256, 0, stream>>>(delta, xs, Bm, Cm, A_log, H0,
                                                          xz, D_par, ycore);
    }

    k_outproj<<<MROWS / 16, 64, 0, stream>>>(ycore, W_out, x, out);
}